// MedianBlur_30966714204228
// MI455X (gfx1250) — compile-verified
//
#include <hip/hip_runtime.h>

// 3x3 median blur + residual blend for x: [8, 3, 1024, 1024] f32.
// out = x + 0.2 * (median3x3_zero_pad(x) - x)
//
// Memory-bound: ~201 MB traffic -> ~8.6 us floor at 23.3 TB/s HBM.
// CDNA5 design: async global->LDS staging (ASYNCcnt path, no VGPR round-trip),
// wave32 blocks, ds_load_b128 window reads, med3 network, NT b128 stores.

typedef float f4 __attribute__((ext_vector_type(4)));

static constexpr int W = 1024;
static constexpr int H = 1024;

static constexpr int BLK_X  = 256;  // output cols per block
static constexpr int BLK_Y  = 16;   // output rows per block
static constexpr int TILE_W = 272;  // staged cols: [bx*256-8, bx*256+263]
static constexpr int TILE_H = 18;   // staged rows: [by*16-1,  by*16+16]
static constexpr int NCHUNK = TILE_H * (TILE_W / 4);  // 1224 b128 chunks

__device__ __forceinline__ float min3f(float a, float b, float c) {
    return fminf(fminf(a, b), c);            // -> v_min3_num_f32
}
__device__ __forceinline__ float max3f(float a, float b, float c) {
    return fmaxf(fmaxf(a, b), c);            // -> v_max3_num_f32
}
__device__ __forceinline__ float med3f(float a, float b, float c) {
    return __builtin_amdgcn_fmed3f(a, b, c); // -> v_med3_num_f32
}

__global__ __launch_bounds__(256) void median3x3_res_kernel(
    const float* __restrict__ in, float* __restrict__ out)
{
    __shared__ __attribute__((aligned(16))) float tile[TILE_H][TILE_W];

    const int t  = threadIdx.x;
    const int tx = t & 63;   // 64 threads across width (x4 cols each)
    const int ty = t >> 6;   // 4 thread-rows (x4 rows each)
    const int bx = blockIdx.x, by = blockIdx.y;
    const int x0 = bx * BLK_X + tx * 4;
    const int y0 = by * BLK_Y + ty * 4;

    const size_t img_off = (size_t)blockIdx.z * (size_t)(W * H);
    const float* img = in + img_off;
    float* oimg = out + img_off;

    // Warm L2 one y-tile ahead (gfx1250 global_prefetch_b8).
    {
        int yp = y0 + BLK_Y; if (yp > H - 1) yp = H - 1;
        __builtin_prefetch(img + (size_t)yp * W + x0, 0, 1);
    }

    // ---- Stage tile into LDS with CDNA5 async copies (ASYNCcnt path). ----
    // 1224 b128 chunks over 256 threads (4 full passes + 1 guarded pass).
    // Global addresses are clamped into the image so every copy is legal;
    // zero padding is applied later via 0/1 scales keyed on true coords.
    const int gx0 = bx * BLK_X - 8;   // global col of LDS col 0
    const int gy0 = by * BLK_Y - 1;   // global row of LDS row 0
    const unsigned lds_base = (unsigned)(uintptr_t)(&tile[0][0]);
#pragma unroll
    for (int i = 0; i < 5; ++i) {
        const int k = t + 256 * i;
        if (k < NCHUNK) {
            const int r  = k / (TILE_W / 4);
            const int cc = k - r * (TILE_W / 4);
            int yr = gy0 + r;
            yr = yr < 0 ? 0 : (yr > H - 1 ? H - 1 : yr);
            int gc = gx0 + cc * 4;
            gc = gc < 0 ? 0 : (gc > W - 4 ? W - 4 : gc);
            const unsigned voff = (unsigned)((yr * W + gc) * 4);
            const unsigned ldso = lds_base + (unsigned)((r * TILE_W + cc * 4) * 4);
            asm volatile("global_load_async_to_lds_b128 %0, %1, %2 offset:0"
                         :: "v"(ldso), "v"(voff), "s"(img)
                         : "memory");
        }
    }
    // Each wave drains its own async copies, then the barrier guarantees the
    // whole tile is visible to all 8 waves.
    asm volatile("s_wait_asynccnt 0" ::: "memory");
    __syncthreads();

    // ---- Read the 6x6 window from LDS (left b32, middle aligned b128, right b32).
    float v[6][6];
    const int lx = tx * 4 + 7;   // LDS col of window-left (global x0-1)
#pragma unroll
    for (int r6 = 0; r6 < 6; ++r6) {
        const int lr = ty * 4 + r6;            // LDS row of global row y0-1+r6
        const int yr = y0 - 1 + r6;
        const float srow = ((unsigned)yr < (unsigned)H) ? 1.0f : 0.0f;
        const float sl = (x0 > 0) ? srow : 0.0f;
        const float sr = (x0 + 4 < W) ? srow : 0.0f;

        const float lv = tile[lr][lx];
        const f4 m = *reinterpret_cast<const f4*>(&tile[lr][lx + 1]); // 16B aligned
        const float rv = tile[lr][lx + 5];

        v[r6][0] = lv * sl;
        v[r6][1] = m.x * srow;
        v[r6][2] = m.y * srow;
        v[r6][3] = m.z * srow;
        v[r6][4] = m.w * srow;
        v[r6][5] = rv * sr;
    }

    // ---- Median of 9 in 13 ops/pixel: sort each column (min/med/max), then
    // med3(max(mins), med(meds), min(maxes)).
#pragma unroll
    for (int yy = 0; yy < 4; ++yy) {
        float lo[6], me[6], hi[6];
#pragma unroll
        for (int j = 0; j < 6; ++j) {
            const float a = v[yy][j], b = v[yy + 1][j], c = v[yy + 2][j];
            lo[j] = min3f(a, b, c);
            me[j] = med3f(a, b, c);
            hi[j] = max3f(a, b, c);
        }
        f4 o;
#pragma unroll
        for (int jj = 0; jj < 4; ++jj) {
            const float mn  = max3f(lo[jj], lo[jj + 1], lo[jj + 2]);
            const float md  = med3f(me[jj], me[jj + 1], me[jj + 2]);
            const float mx  = min3f(hi[jj], hi[jj + 1], hi[jj + 2]);
            const float med = med3f(mn, md, mx);
            const float x   = v[yy + 1][jj + 1];
            o[jj] = fmaf(0.2f, med - x, x);   // x + 0.2*(med - x)
        }
        float* orow = oimg + (size_t)(y0 + yy) * W + x0;
        // NT store: keep output's 96MB from evicting input rows out of L2.
        __builtin_nontemporal_store(o, reinterpret_cast<f4*>(orow));
    }
}

extern "C" void kernel_launch(void* const* d_in, const int* in_sizes, int n_in,
                              void* d_out, int out_size, void* d_ws, size_t ws_size,
                              hipStream_t stream) {
    (void)in_sizes; (void)n_in; (void)d_ws; (void)ws_size; (void)out_size;
    const float* x = (const float*)d_in[0];
    float* out = (float*)d_out;

    // [8,3,1024,1024] -> 24 independent 1024x1024 images.
    dim3 grid(W / BLK_X, H / BLK_Y, 24);
    dim3 block(256);
    median3x3_res_kernel<<<grid, block, 0, stream>>>(x, out);
}